// Bedroom_54357106098316
// MI455X (gfx1250) — compile-verified
//
#include <hip/hip_runtime.h>

// ---------------------------------------------------------------------------
// Full 1-D convolution (data ⊛ rir) + peak normalization, via f32 WMMA.
//
//   y[j] = sum_k h[k] * x[j-k],  j = 0 .. N+K-2
//
// Tile of 256 outputs per wave: D[m,n] = y[j0 + m + 16n].
//   A[m,s] = h[4r + m - s]           (16x4 Toeplitz slice of the filter, LDS)
//   B[s,n] = x[j0 - 4r + s + 16n]    (4x16 slice of the signal, L2-resident)
//   D += A x B  covers taps k in [4r+m-3, 4r+m]; r sweep covers all K taps.
// Filter staged through LDS in 8 chunks of ~16KB; signal read directly
// (sliding 1KB window, ~64x reuse in WGP$/L2).
// ---------------------------------------------------------------------------

typedef float v2f __attribute__((ext_vector_type(2)));
typedef float v4f __attribute__((ext_vector_type(4)));
typedef float v8f __attribute__((ext_vector_type(8)));

#define N_SAMPLES 2097152
#define RIR_LEN   32000
#define OUT_LEN   (N_SAMPLES + RIR_LEN - 1)     // 2129151
#define XOFF      32768                         // left zero pad of signal
#define LXP       (XOFF + N_SAMPLES + 32768)    // padded signal length
#define HOFF      8                             // left zero pad of filter
#define LHP       32768                         // padded filter length
#define NTILES    ((OUT_LEN + 255) / 256)       // 8317 output tiles
#define CHUNKS    8                             // filter chunks through LDS
#define CHITER    1008                          // WMMA steps per chunk
#define NITER     (CHUNKS * CHITER)             // 8064 >= K/4+1; extras hit 0-pad
#define HSPAN     (4 * (CHITER - 1) + 19)       // 4047 filter floats per chunk
#define HLDS      4048                          // padded LDS chunk size

// ---------------------------------------------------------------------------
// Pass 0: zero-pad signal & filter into workspace; clear max accumulator.
// ---------------------------------------------------------------------------
__global__ void __launch_bounds__(256)
prep_kernel(const float* __restrict__ x, const float* __restrict__ h,
            float* __restrict__ xp, float* __restrict__ hp,
            unsigned int* __restrict__ maxbits)
{
    int i = blockIdx.x * 256 + threadIdx.x;
    if (i == 0) *maxbits = 0u;                       // |y| >= 0 -> uint-max works
    if (i < LXP) {
        int k = i - XOFF;
        xp[i] = (k >= 0 && k < N_SAMPLES) ? x[k] : 0.0f;
    }
    if (i < LHP) {
        int k = i - HOFF;
        hp[i] = (k >= 0 && k < RIR_LEN) ? h[k] : 0.0f;
    }
}

// ---------------------------------------------------------------------------
// Pass 1: one wave32 per 256-output tile; V_WMMA_F32_16X16X4_F32 inner loop,
// filter chunks staged through LDS (A reads become conflict-free ds_read2).
// ---------------------------------------------------------------------------
__global__ void __launch_bounds__(256)
conv_wmma_kernel(const float* __restrict__ xp, const float* __restrict__ hp,
                 float* __restrict__ out, unsigned int* __restrict__ maxbits)
{
    __shared__ float sh[HLDS];

    const int lane = threadIdx.x & 31;
    const int wave = threadIdx.x >> 5;
    const int tile = blockIdx.x * 8 + wave;
    // Early-exit waves are safe: HW barrier completes on signal-or-terminate.
    if (tile >= NTILES) return;

    const int j0 = tile * 256;
    const int ln = lane & 15;                    // M for A, N for B/C/D
    const int kh = lane >> 4;                    // lane half: K pair / M half

    // B per-lane base: x[j0 - 4r + 2*kh + 16*ln] (+1)  -> one aligned b64 load
    const float* bq = xp + (XOFF + j0 + 2 * kh + 16 * ln);
    // A per-lane LDS base: sh[local(4r + ln - 2kh - 1)] and next element
    const float* aq = sh + (ln - 2 * kh + 2);

    v8f c = {0.f, 0.f, 0.f, 0.f, 0.f, 0.f, 0.f, 0.f};

    int hbase = HOFF - 3;                        // global hp index of sh[0]
    for (int ch = 0; ch < CHUNKS; ++ch) {
        __syncthreads();                         // prev chunk fully consumed
        for (int i = threadIdx.x; i < HSPAN; i += 256)
            sh[i] = hp[hbase + i];               // stage filter chunk -> LDS
        __syncthreads();

        const float* bq0 = bq - 4 * CHITER * ch; // B base for this chunk
#pragma unroll 8
        for (int rr = 0; rr < CHITER; ++rr) {
            v2f b = *(const v2f*)(bq0 - 4 * rr); // x[c0+s+16n], s = 2kh,2kh+1
            v2f a;
            a.y = aq[4 * rr];                    // h[4r + m - 2kh - 1] (s=2kh+1)
            a.x = aq[4 * rr + 1];                // h[4r + m - 2kh]     (s=2kh)
            // D = A x B + C   (16x16x4 f32, exact f32 accumulation)
            c = __builtin_amdgcn_wmma_f32_16x16x4_f32(
                    /*neg_a=*/false, a, /*neg_b=*/false, b,
                    /*c_mod=*/(short)0, c, /*reuse_a=*/false, /*reuse_b=*/false);
        }
        hbase += 4 * CHITER;
    }

    // D layout: c[v] = y[j0 + (v + 8*kh) + 16*ln] -> 8 consecutive outputs/lane
    const int base = j0 + 16 * ln + 8 * kh;
    float m = 0.0f;
    if (j0 + 256 <= OUT_LEN) {                   // full tile: two b128 stores
        v4f lo = {c[0], c[1], c[2], c[3]};
        v4f hi = {c[4], c[5], c[6], c[7]};
        *(v4f*)(out + base)     = lo;
        *(v4f*)(out + base + 4) = hi;
#pragma unroll
        for (int v = 0; v < 8; ++v) m = fmaxf(m, fabsf(c[v]));
    } else {                                     // last tile: guarded tail
#pragma unroll
        for (int v = 0; v < 8; ++v) {
            int j = base + v;
            if (j < OUT_LEN) {
                out[j] = c[v];
                m = fmaxf(m, fabsf(c[v]));
            }
        }
    }

    // wave32 max-reduce, then one atomic per wave
    for (int off = 16; off > 0; off >>= 1)
        m = fmaxf(m, __shfl_xor(m, off, 32));
    if (lane == 0)
        atomicMax(maxbits, __float_as_uint(m));
}

// ---------------------------------------------------------------------------
// Pass 2: y /= max(|y|) iff max > 1  (d_out fully rewritten by pass 1 every
// call, so this in-place scale is graph-replay safe).
// ---------------------------------------------------------------------------
__global__ void __launch_bounds__(256)
scale_kernel(float* __restrict__ out, const unsigned int* __restrict__ maxbits)
{
    int i = blockIdx.x * 256 + threadIdx.x;
    if (i < OUT_LEN) {
        float mval = __uint_as_float(*maxbits);
        if (mval > 1.0f) out[i] = out[i] / mval;
    }
}

// ---------------------------------------------------------------------------
// Launch: d_in[0] = data (f32, N), d_in[1] = rir (f32, K), d_out = y (f32).
// Workspace (floats): [0] max bits | [256, 256+LXP) padded x | then padded h.
// Requires ~8.4 MB of d_ws.
// ---------------------------------------------------------------------------
extern "C" void kernel_launch(void* const* d_in, const int* in_sizes, int n_in,
                              void* d_out, int out_size, void* d_ws, size_t ws_size,
                              hipStream_t stream) {
    const float* x  = (const float*)d_in[0];
    const float* h  = (const float*)d_in[1];
    float*       out = (float*)d_out;

    float*        ws      = (float*)d_ws;
    unsigned int* maxbits = (unsigned int*)ws;
    float*        xp      = ws + 256;
    float*        hp      = xp + LXP;

    (void)in_sizes; (void)n_in; (void)out_size; (void)ws_size;

    const int prepBlocks = (LXP + 255) / 256;
    prep_kernel<<<prepBlocks, 256, 0, stream>>>(x, h, xp, hp, maxbits);

    const int convBlocks = (NTILES + 7) / 8;     // 8 waves (tiles) per block
    conv_wmma_kernel<<<convBlocks, 256, 0, stream>>>(xp, hp, out, maxbits);

    const int scaleBlocks = (OUT_LEN + 255) / 256;
    scale_kernel<<<scaleBlocks, 256, 0, stream>>>(out, maxbits);
}